// GATBlock_69389491634795
// MI455X (gfx1250) — compile-verified
//
#include <hip/hip_runtime.h>
#include <hip/hip_bf16.h>

typedef __attribute__((ext_vector_type(2))) float v2f;
typedef __attribute__((ext_vector_type(8))) float v8f;

#define LN_EPS    1e-5f
#define NEG_SLOPE 0.2f

// ---------------------------------------------------------------------------
// float atomic max via sign-split integer atomics (monotone float<->int map).
// Works for all non-NaN floats, including the -inf initial value.
// ---------------------------------------------------------------------------
__device__ __forceinline__ void atomic_max_f32(float* addr, float val) {
  if (val >= 0.0f) {
    atomicMax((int*)addr, __float_as_int(val));
  } else {
    atomicMin((unsigned int*)addr, __float_as_uint(val));
  }
}

__device__ __forceinline__ float gelu_exact(float g) {
  return 0.5f * g * (1.0f + erff(g * 0.70710678118654752440f));
}

// ---------------------------------------------------------------------------
// K0: init accumulator to 0, emax to -inf, denom to 0
// ---------------------------------------------------------------------------
__global__ void k_init(float* __restrict__ out, float* __restrict__ emax,
                       float* __restrict__ denom, int nOut, int nAtt) {
  int i = blockIdx.x * blockDim.x + threadIdx.x;
  if (i < nOut) out[i] = 0.0f;
  if (i < nAtt) { emax[i] = -__builtin_inff(); denom[i] = 0.0f; }
}

// ---------------------------------------------------------------------------
// K1: H = X @ W  (M x 128) = (M x 128)(128 x 128), exact fp32 via
//     V_WMMA_F32_16X16X4_F32. One wave owns a 16x16 C tile; a 256-thread
//     block (8 waves) covers one 16-row stripe across all 128 columns.
//
// NOTE: assumes M % 16 == 0 (problem fixes N_NODES = 50000 = 3125*16), so
// there is no M-tail: loads and stores are unguarded straight-line code and
// EXEC stays all-ones around the WMMAs (ISA requirement).
//
// A-frag (16x4 f32):  lane L holds A[m= L%16][k + v + 2*(L/16)], v=0,1
// B-frag (4x16 f32):  lane L holds B[k + v + 2*(L/16)][n= L%16], v=0,1
// C/D  (16x16 f32):   VGPR v, lane L -> row v + 8*(L/16), col L%16
// ---------------------------------------------------------------------------
__global__ void __launch_bounds__(256) k_gemm(const float* __restrict__ X,
                                              const float* __restrict__ W,
                                              float* __restrict__ H, int M) {
  const int lane  = threadIdx.x & 31;
  const int wave  = threadIdx.x >> 5;     // N-tile 0..7
  const int half  = lane >> 4;            // 0 or 1
  const int l15   = lane & 15;
  const int mtile = blockIdx.x;

  const int m = mtile * 16 + l15;         // M % 16 == 0 -> always in range
  const int n = wave * 16 + l15;          // 0..127

  const float* __restrict__ Xp = X + (size_t)m * 128;

  v8f acc = {};
#pragma unroll 4
  for (int k = 0; k < 128; k += 4) {
    const int ka = k + 2 * half;          // even -> 8B-aligned v2f load
    v2f a = *(const v2f*)(Xp + ka);
    v2f b;
    b.x = W[(size_t)(ka + 0) * 128 + n];
    b.y = W[(size_t)(ka + 1) * 128 + n];
    acc = __builtin_amdgcn_wmma_f32_16x16x4_f32(
        /*neg_a=*/false, a, /*neg_b=*/false, b,
        /*c_mod=*/(short)0, acc, /*reuse_a=*/false, /*reuse_b=*/false);
  }

  // Unguarded epilogue: 8 straight-line stores, rows strided by 128 floats.
  float* __restrict__ Hp = H + ((size_t)(mtile * 16 + 8 * half)) * 128 + n;
#pragma unroll
  for (int v = 0; v < 8; ++v) {
    Hp[(size_t)v * 128] = acc[v];
  }
}

// ---------------------------------------------------------------------------
// K2: per-node attention scores. One wave per node; lane owns 4 channels;
//     8 lanes form one head (32 channels); width-8 butterfly reduction.
//     att_src/att_dst are [H,C] flat = 128 floats, so flat index == channel.
// ---------------------------------------------------------------------------
__global__ void k_scores(const float* __restrict__ H,
                         const float* __restrict__ att_src,
                         const float* __restrict__ att_dst,
                         float* __restrict__ asrc, float* __restrict__ adst,
                         int N) {
  int gtid = blockIdx.x * blockDim.x + threadIdx.x;
  int node = gtid >> 5;
  int lane = gtid & 31;
  if (node >= N) return;
  const int c0 = lane * 4;
  const float4 hv = *(const float4*)(H + (size_t)node * 128 + c0);
  const float4 as = *(const float4*)(att_src + c0);
  const float4 ad = *(const float4*)(att_dst + c0);
  float ps = hv.x * as.x + hv.y * as.y + hv.z * as.z + hv.w * as.w;
  float pd = hv.x * ad.x + hv.y * ad.y + hv.z * ad.z + hv.w * ad.w;
#pragma unroll
  for (int m = 1; m < 8; m <<= 1) {
    ps += __shfl_xor(ps, m, 32);
    pd += __shfl_xor(pd, m, 32);
  }
  if ((lane & 7) == 0) {
    const int head = lane >> 3;
    asrc[(size_t)node * 4 + head] = ps;
    adst[(size_t)node * 4 + head] = pd;
  }
}

// ---------------------------------------------------------------------------
// K3/K4: edge passes over E real edges + N self-loops (i>=E => src=dst=i-E).
//        One thread per (edge, head).
// ---------------------------------------------------------------------------
__global__ void k_edge_max(const int* __restrict__ srcs, const int* __restrict__ dsts,
                           const float* __restrict__ asrc, const float* __restrict__ adst,
                           float* __restrict__ emax, int E, int total) {
  int tid = blockIdx.x * blockDim.x + threadIdx.x;
  int i = tid >> 2, h = tid & 3;
  if (i >= total) return;
  const int s = (i < E) ? srcs[i] : (i - E);
  const int d = (i < E) ? dsts[i] : (i - E);
  float e = asrc[(size_t)s * 4 + h] + adst[(size_t)d * 4 + h];
  e = (e > 0.0f) ? e : NEG_SLOPE * e;
  atomic_max_f32(&emax[(size_t)d * 4 + h], e);
}

__global__ void k_edge_sum(const int* __restrict__ srcs, const int* __restrict__ dsts,
                           const float* __restrict__ asrc, const float* __restrict__ adst,
                           const float* __restrict__ emax, float* __restrict__ denom,
                           int E, int total) {
  int tid = blockIdx.x * blockDim.x + threadIdx.x;
  int i = tid >> 2, h = tid & 3;
  if (i >= total) return;
  const int s = (i < E) ? srcs[i] : (i - E);
  const int d = (i < E) ? dsts[i] : (i - E);
  float e = asrc[(size_t)s * 4 + h] + adst[(size_t)d * 4 + h];
  e = (e > 0.0f) ? e : NEG_SLOPE * e;
  const float ex = expf(e - emax[(size_t)d * 4 + h]);
  atomicAdd(&denom[(size_t)d * 4 + h], ex);
}

// ---------------------------------------------------------------------------
// K5: message scatter. One wave per edge; lane owns 4 channels (float4
//     coalesced gather of h[src], 512B per wave), 4 f32 atomic adds into
//     the L2-resident accumulator.
// ---------------------------------------------------------------------------
__global__ void k_scatter(const int* __restrict__ srcs, const int* __restrict__ dsts,
                          const float* __restrict__ H, const float* __restrict__ asrc,
                          const float* __restrict__ adst, const float* __restrict__ emax,
                          const float* __restrict__ denom, float* __restrict__ out,
                          int E, int total) {
  unsigned tid = blockIdx.x * blockDim.x + threadIdx.x;
  unsigned i = tid >> 5;
  const int part = tid & 31;
  if (i >= (unsigned)total) return;
  const int s = (i < (unsigned)E) ? srcs[i] : (int)(i - (unsigned)E);
  const int d = (i < (unsigned)E) ? dsts[i] : (int)(i - (unsigned)E);
  const int h = part >> 3;
  float e = asrc[(size_t)s * 4 + h] + adst[(size_t)d * 4 + h];
  e = (e > 0.0f) ? e : NEG_SLOPE * e;
  const float alpha = expf(e - emax[(size_t)d * 4 + h]) / denom[(size_t)d * 4 + h];
  const float4 hv = *(const float4*)(H + (size_t)s * 128 + part * 4);
  float* o = out + (size_t)d * 128 + part * 4;
  atomicAdd(o + 0, hv.x * alpha);
  atomicAdd(o + 1, hv.y * alpha);
  atomicAdd(o + 2, hv.z * alpha);
  atomicAdd(o + 3, hv.w * alpha);
}

// ---------------------------------------------------------------------------
// K6: +bias, LayerNorm (full-wave butterfly), residual, exact GELU.
//     One wave per node; reads & writes d_out in place (lane-exclusive).
// ---------------------------------------------------------------------------
__global__ void k_finalize(const float* __restrict__ acc, const float* __restrict__ x,
                           const float* __restrict__ bias, const float* __restrict__ gamma,
                           const float* __restrict__ beta, float* __restrict__ out, int N) {
  int gtid = blockIdx.x * blockDim.x + threadIdx.x;
  int node = gtid >> 5;
  int lane = gtid & 31;
  if (node >= N) return;
  const int c0 = lane * 4;
  float4 v = *(const float4*)(acc + (size_t)node * 128 + c0);
  const float4 bb = *(const float4*)(bias + c0);
  v.x += bb.x; v.y += bb.y; v.z += bb.z; v.w += bb.w;

  float s = v.x + v.y + v.z + v.w;
#pragma unroll
  for (int m = 1; m < 32; m <<= 1) s += __shfl_xor(s, m, 32);
  const float mu = s * (1.0f / 128.0f);

  const float dx = v.x - mu, dy = v.y - mu, dz = v.z - mu, dw = v.w - mu;
  float q = dx * dx + dy * dy + dz * dz + dw * dw;
#pragma unroll
  for (int m = 1; m < 32; m <<= 1) q += __shfl_xor(q, m, 32);
  const float inv = 1.0f / sqrtf(q * (1.0f / 128.0f) + LN_EPS);

  const float4 g  = *(const float4*)(gamma + c0);
  const float4 b2 = *(const float4*)(beta + c0);
  const float4 xv = *(const float4*)(x + (size_t)node * 128 + c0);
  float4 r;
  r.x = gelu_exact(dx * inv * g.x + b2.x + xv.x);
  r.y = gelu_exact(dy * inv * g.y + b2.y + xv.y);
  r.z = gelu_exact(dz * inv * g.z + b2.z + xv.z);
  r.w = gelu_exact(dw * inv * g.w + b2.w + xv.w);
  *(float4*)(out + (size_t)node * 128 + c0) = r;
}

// ---------------------------------------------------------------------------
extern "C" void kernel_launch(void* const* d_in, const int* in_sizes, int n_in,
                              void* d_out, int out_size, void* d_ws, size_t ws_size,
                              hipStream_t stream) {
  const float* x       = (const float*)d_in[0];
  const int*   ei      = (const int*)  d_in[1];   // [2, E] flat
  const float* W       = (const float*)d_in[2];
  const float* att_src = (const float*)d_in[3];
  const float* att_dst = (const float*)d_in[4];
  const float* bias    = (const float*)d_in[5];
  const float* gamma   = (const float*)d_in[6];
  const float* beta    = (const float*)d_in[7];
  float* out = (float*)d_out;

  const int N = in_sizes[0] / 128;
  const int E = in_sizes[1] / 2;
  const int total = E + N;                 // edges + self-loops

  // workspace carve-out: H | asrc | adst | emax | denom  (~28.8 MB)
  float* H     = (float*)d_ws;
  float* asrc  = H     + (size_t)N * 128;
  float* adst  = asrc  + (size_t)N * 4;
  float* emax  = adst  + (size_t)N * 4;
  float* denom = emax  + (size_t)N * 4;

  const int* srcs = ei;
  const int* dsts = ei + E;

  const int nOut = N * 128;
  k_init<<<(nOut + 255) / 256, 256, 0, stream>>>(out, emax, denom, nOut, N * 4);

  k_gemm<<<N / 16, 256, 0, stream>>>(x, W, H, N);   // N % 16 == 0 (50000)

  k_scores<<<(N * 32 + 255) / 256, 256, 0, stream>>>(H, att_src, att_dst, asrc, adst, N);

  const int tMax = total * 4;
  k_edge_max<<<(tMax + 255) / 256, 256, 0, stream>>>(srcs, dsts, asrc, adst, emax, E, total);
  k_edge_sum<<<(tMax + 255) / 256, 256, 0, stream>>>(srcs, dsts, asrc, adst, emax, denom, E, total);

  const long long tSc = (long long)total * 32;
  k_scatter<<<(unsigned)((tSc + 255) / 256), 256, 0, stream>>>(srcs, dsts, H, asrc, adst,
                                                               emax, denom, out, E, total);

  k_finalize<<<(N * 32 + 255) / 256, 256, 0, stream>>>(out, x, bias, gamma, beta, out, N);
}